// MultiScaleHead_51677046505533
// MI455X (gfx1250) — compile-verified
//
#include <hip/hip_runtime.h>
#include <hip/hip_bf16.h>

// Problem constants (from reference)
#define B_  16
#define L_  2048
#define H_  1024
#define P_  16
#define S_  128

typedef __attribute__((ext_vector_type(2))) float v2f;
typedef __attribute__((ext_vector_type(8))) float v8f;

// ---------------------------------------------------------------------------
// Phase 1: build the fused coefficient matrix
//   A'[b,p,s] = par_valid[b,p] * sel[b,p,s] / (sent_cnt[b,p] * par_cnt[b])
// so that  final[b,h] = sum_{p,s} A'[b,p,s] * emb[b,s,h]
// One block per batch, 128 threads (thread = sentence s).
// ---------------------------------------------------------------------------
__global__ __launch_bounds__(S_) void msh_phase1(const int* __restrict__ sent_head,
                                                 const int* __restrict__ par_head,
                                                 const int* __restrict__ par_tail,
                                                 float* __restrict__ Ap) {
  const int b = blockIdx.x;
  const int s = threadIdx.x;            // 0..127

  __shared__ int   s_ph[P_], s_pt[P_], s_any[P_];
  __shared__ float s_cnt[P_], s_w[P_];
  __shared__ int   s_parcnt;
  __shared__ float s_invpar;

  if (s < P_) {
    s_ph[s]  = par_head[b * P_ + s];
    s_pt[s]  = par_tail[b * P_ + s];
    s_cnt[s] = 0.0f;
    s_any[s] = 0;
  }
  if (s == 0) s_parcnt = 0;
  __syncthreads();

  const int sh = sent_head[b * S_ + s];
  unsigned selbits = 0u;
#pragma unroll
  for (int p = 0; p < P_; ++p) {
    const bool sel = (s_ph[p] <= sh) && (sh <= s_pt[p]);
    if (sel) {
      selbits |= (1u << p);
      atomicAdd(&s_cnt[p], 1.0f);
      if (sh != 0) atomicOr(&s_any[p], 1);
    }
  }
  __syncthreads();

  if (s < P_) {
    const int  p      = s;
    const bool pvalid = ((s_pt[p] - s_ph[p]) > 2) && (s_any[p] != 0);
    const float cnt   = fmaxf(s_cnt[p], 1.0f);
    s_w[p] = pvalid ? (1.0f / cnt) : 0.0f;
    if (pvalid) atomicAdd(&s_parcnt, 1);
  }
  __syncthreads();
  if (s == 0) s_invpar = 1.0f / fmaxf((float)s_parcnt, 1.0f);
  __syncthreads();

  const float invpar = s_invpar;
#pragma unroll
  for (int p = 0; p < P_; ++p) {
    const float v = ((selbits >> p) & 1u) ? (s_w[p] * invpar) : 0.0f;
    Ap[((size_t)b * P_ + p) * S_ + s] = v;
  }
}

// ---------------------------------------------------------------------------
// Phase 2: per (batch, 64-column chunk):
//   - gather the 128 sentence rows straight into LDS with
//     GLOBAL_LOAD_ASYNC_TO_LDS_B128 (CDNA5 direct global->LDS path, ASYNCcnt)
//   - 4 waves each compute a 16x16 tile of A'(16x128) @ emb(128x64) via 32
//     chained V_WMMA_F32_16X16X4_F32 ops, column-sum the tile, write final.
// Block = 128 threads (4 wave32). EXEC is all-ones at every WMMA / async op.
// ---------------------------------------------------------------------------
#define NCHUNK    64                 // columns of H per workgroup
#define LE_STRIDE 68                 // padded row stride: 272B = 17*16 (16B aligned, kills bank conflicts)
#define LA_STRIDE 130                // padded A' stride (non-multiple of 64 banks)

__global__ __launch_bounds__(128) void msh_phase2(const float* __restrict__ backbone,
                                                  const int* __restrict__ sent_head,
                                                  const float* __restrict__ Ap,
                                                  float* __restrict__ finalbuf) {
  const int b  = blockIdx.x;         // batch
  const int h0 = blockIdx.y * NCHUNK;

  __shared__ __attribute__((aligned(16))) float lE[S_ * LE_STRIDE]; // gathered emb chunk
  __shared__ float lA[P_ * LA_STRIDE];                              // A' coefficients

  const int tid = threadIdx.x;       // 0..127

  // Async-gather emb rows directly into LDS: 128 rows * 16 xfers of 16B each.
  // 2048 b128 transfers / 128 threads = 16 per lane, uniform trip count.
  for (int i = tid; i < S_ * (NCHUNK / 4); i += 128) {
    const int row = i >> 4;          // sentence index
    const int q   = i & 15;          // 16B chunk within the 64-col slice
    const int sh  = sent_head[b * S_ + row];
    const float* gptr = backbone + (((size_t)b * L_ + sh) * H_ + h0) + 4 * q;
    // Low 32 bits of a generic LDS pointer == LDS byte address (aperture is in [63:32]).
    const unsigned ldsoff = (unsigned)(size_t)(&lE[row * LE_STRIDE + 4 * q]);
    asm volatile("global_load_async_to_lds_b128 %0, %1, off"
                 :
                 : "v"(ldsoff), "v"(gptr)
                 : "memory");
  }

  // Stage A' (16x128) with padded stride (small; regular stores).
  for (int i = tid; i < P_ * S_; i += 128) {
    lA[(i >> 7) * LA_STRIDE + (i & 127)] = Ap[(size_t)b * P_ * S_ + i];
  }

  // Drain this wave's async copies, then make LDS visible workgroup-wide.
  asm volatile("s_wait_asynccnt 0" ::: "memory");
  __syncthreads();

  const int wave = tid >> 5;         // 0..3 -> 16-column tile
  const int l    = tid & 31;         // lane
  const int n0   = wave * 16;
  const int m    = l & 15;           // row-in-half (A: M, B/D: N offset)
  const bool lo  = (l < 16);

  v8f c = {};
  for (int k0 = 0; k0 < S_; k0 += 4) {
    // A 16x4 fragment (ISA layout): lanes 0-15: {K=k0,K=k0+1}; lanes 16-31: {K=k0+2,K=k0+3}; M = l&15
    const int ka = k0 + (lo ? 0 : 2);
    v2f a;
    a.x = lA[m * LA_STRIDE + ka];
    a.y = lA[m * LA_STRIDE + ka + 1];
    // B 4x16 fragment: rows striped across lane halves, K contiguous across VGPRs:
    //   V0: lanes0-15 row k0+0, lanes16-31 row k0+1 ; V1: rows k0+2 / k0+3 ; N = n0 + (l&15)
    const int kb = k0 + (lo ? 0 : 1);
    v2f bb;
    bb.x = lE[kb * LE_STRIDE + n0 + m];
    bb.y = lE[(kb + 2) * LE_STRIDE + n0 + m];
    // D = A*B + C  (8 args: neg_a, A, neg_b, B, c_mod, C, reuse_a, reuse_b)
    c = __builtin_amdgcn_wmma_f32_16x16x4_f32(false, a, false, bb, (short)0, c,
                                              false, false);
  }

  // Column-sum the 16x16 tile over M (paragraphs).
  // Lane l (<16): VGPR r holds M=r, N=l ; lane l+16: M=8+r, N=l.
  float colsum = c[0] + c[1] + c[2] + c[3] + c[4] + c[5] + c[6] + c[7];
  colsum += __shfl_xor(colsum, 16, 32);
  if (l < 16) {
    finalbuf[(size_t)b * H_ + h0 + n0 + l] = colsum;
  }
}

// ---------------------------------------------------------------------------
// Phase 3: out[b, 0..4] = final[b,:] @ W.T + bias  (deterministic tree reduce)
// ---------------------------------------------------------------------------
__global__ __launch_bounds__(256) void msh_phase3(const float* __restrict__ finalbuf,
                                                  const float* __restrict__ W,
                                                  const float* __restrict__ bias,
                                                  float* __restrict__ out) {
  const int b   = blockIdx.x;
  const int tid = threadIdx.x;       // 0..255

  float acc[5] = {0.f, 0.f, 0.f, 0.f, 0.f};
  for (int h = tid; h < H_; h += 256) {
    const float f = finalbuf[(size_t)b * H_ + h];
#pragma unroll
    for (int cc = 0; cc < 5; ++cc) acc[cc] += f * W[cc * H_ + h];
  }

  __shared__ float red[5][256];
#pragma unroll
  for (int cc = 0; cc < 5; ++cc) red[cc][tid] = acc[cc];
  __syncthreads();
  for (int s = 128; s > 0; s >>= 1) {
    if (tid < s) {
#pragma unroll
      for (int cc = 0; cc < 5; ++cc) red[cc][tid] += red[cc][tid + s];
    }
    __syncthreads();
  }
  if (tid < 5) out[b * 5 + tid] = red[tid][0] + bias[tid];
}

// ---------------------------------------------------------------------------
extern "C" void kernel_launch(void* const* d_in, const int* in_sizes, int n_in,
                              void* d_out, int out_size, void* d_ws, size_t ws_size,
                              hipStream_t stream) {
  (void)in_sizes; (void)n_in; (void)out_size; (void)ws_size;

  const float* backbone  = (const float*)d_in[0];   // (B,L,H) f32
  // d_in[1] attention_mask: unused by reference
  const int*   par_head  = (const int*)d_in[2];     // (B,P)
  const int*   par_tail  = (const int*)d_in[3];     // (B,P)
  // d_in[4] paragraph_attention_mask: unused
  const int*   sent_head = (const int*)d_in[5];     // (B,S)
  // d_in[6] sentence_tail_idxs, d_in[7] sentence_attention_mask: unused
  const float* W         = (const float*)d_in[8];   // (5,H)
  const float* bias      = (const float*)d_in[9];   // (5,)
  float*       out       = (float*)d_out;           // (B,5)

  float* Ap       = (float*)d_ws;                   // B*P*S floats  (128 KB)
  float* finalbuf = Ap + (size_t)B_ * P_ * S_;      // B*H floats    ( 64 KB)

  msh_phase1<<<dim3(B_), dim3(S_), 0, stream>>>(sent_head, par_head, par_tail, Ap);
  msh_phase2<<<dim3(B_, H_ / NCHUNK), dim3(128), 0, stream>>>(backbone, sent_head, Ap, finalbuf);
  msh_phase3<<<dim3(B_), dim3(256), 0, stream>>>(finalbuf, W, bias, out);
}